// SAPDTarget_71236327571651
// MI455X (gfx1250) — compile-verified
//
#include <hip/hip_runtime.h>

#define NUM_CLASSES 80
#define NPIX_IMG 21824
#define CLS_CH 82
#define REG_CH 6
#define NBOX 64
#define NIMG 8

__global__ __launch_bounds__(256) void sapd_level_kernel(
    const float* __restrict__ boxes,   // (8,64,5)
    const float* __restrict__ meta,    // (8,64,5)
    float* __restrict__ cls_out,       // (8,21824,82)
    float* __restrict__ regr_out,      // (8,21824,6)
    int lvl, int fh, int fw, float stride, int lvlOff)
{
    const int tid    = threadIdx.x;
    const int b      = blockIdx.y;
    const int block0 = blockIdx.x * 256;
    const int npix   = fh * fw;
    const int P      = (npix - block0) < 256 ? (npix - block0) : 256;

    __shared__ __align__(16) float rawBoxes[NBOX * 5];
    __shared__ float rawMeta[NBOX];
    __shared__ float bx1[NBOX], by1[NBOX], bx2[NBOX], by2[NBOX];
    __shared__ int   qx1[NBOX], qy1[NBOX], qx2[NBOX], qy2[NBOX];
    __shared__ int   blab[NBOX];
    __shared__ unsigned bvld[NBOX];
    __shared__ float res[256][8];

    // ---- Stage box table + per-level meta weights into LDS via CDNA5 async copies
    {
        const float* gBoxes = boxes + (size_t)b * (NBOX * 5);
        if (tid < 80) {  // 80 lanes x 16B = 320 floats
            const float* g = gBoxes + tid * 4;
            unsigned l = (unsigned)(size_t)(&rawBoxes[tid * 4]);
            asm volatile("global_load_async_to_lds_b128 %0, %1, off"
                         :: "v"(l), "v"(g) : "memory");
        }
        if (tid < NBOX) {  // strided gather of meta[b, n, lvl]
            const float* g = meta + ((size_t)b * NBOX + tid) * 5 + lvl;
            unsigned l = (unsigned)(size_t)(&rawMeta[tid]);
            asm volatile("global_load_async_to_lds_b32 %0, %1, off"
                         :: "v"(l), "v"(g) : "memory");
        }
        asm volatile("s_wait_asynccnt 0" ::: "memory");
    }
    __syncthreads();

    // ---- Per-box derived data (matches reference op order exactly)
    if (tid < NBOX) {
        float x1 = rawBoxes[tid * 5 + 0];
        float y1 = rawBoxes[tid * 5 + 1];
        float x2 = rawBoxes[tid * 5 + 2];
        float y2 = rawBoxes[tid * 5 + 3];
        float lf = rawBoxes[tid * 5 + 4];
        bx1[tid] = x1; by1[tid] = y1; bx2[tid] = x2; by2[tid] = y2;
        bvld[tid] = (fabsf(x1) + fabsf(y1) + fabsf(x2) + fabsf(y2)) > 0.0f ? 1u : 0u;
        int lab = (int)lf;
        lab = lab < 0 ? 0 : (lab > NUM_CLASSES - 1 ? NUM_CLASSES - 1 : lab);
        blab[tid] = lab;
        float cx = (x1 + x2) * 0.5f, cy = (y1 + y2) * 0.5f;
        float hw = (x2 - x1) * 0.2f * 0.5f;
        float hh = (y2 - y1) * 0.2f * 0.5f;
        int px1 = (int)floorf((cx - hw) / stride);
        px1 = px1 < 0 ? 0 : (px1 > fw - 1 ? fw - 1 : px1);
        int py1 = (int)floorf((cy - hh) / stride);
        py1 = py1 < 0 ? 0 : (py1 > fh - 1 ? fh - 1 : py1);
        int px2 = (int)ceilf((cx + hw) / stride);
        px2 = px2 < 1 ? 1 : (px2 > fw ? fw : px2);
        px2 = px2 > px1 + 1 ? px2 : px1 + 1;
        int py2 = (int)ceilf((cy + hh) / stride);
        py2 = py2 < 1 ? 1 : (py2 > fh ? fh : py2);
        py2 = py2 > py1 + 1 ? py2 : py1 + 1;
        qx1[tid] = px1; qy1[tid] = py1; qx2[tid] = px2; qy2[tid] = py2;
    }
    __syncthreads();

    // ---- Per-pixel argmin over boxes + winner decode
    if (tid < P) {
        const int p = block0 + tid;
        const int y = p / fw;
        const int x = p - y * fw;
        const float xs = ((float)x + 0.5f) * stride;
        const float ys = ((float)y + 0.5f) * stride;

        float best = 1.0e7f;
        int   win  = 0;
        for (int n = 0; n < NBOX; ++n) {
            bool inside = bvld[n] && (x >= qx1[n]) && (x < qx2[n]) &&
                          (y >= qy1[n]) && (y < qy2[n]);
            float dl = fmaxf(xs - bx1[n], 0.0f);
            float dr = fmaxf(bx2[n] - xs, 0.0f);
            float dt = fmaxf(ys - by1[n], 0.0f);
            float db = fmaxf(by2[n] - ys, 0.0f);
            float area = inside ? (dl + dr) * (dt + db) : 1.0e7f;
            if (area < best) { best = area; win = n; }  // first-occurrence argmin
        }

        float dl = fmaxf(xs - bx1[win], 0.0f);
        float dr = fmaxf(bx2[win] - xs, 0.0f);
        float dt = fmaxf(ys - by1[win], 0.0f);
        float db = fmaxf(by2[win] - ys, 0.0f);
        bool inw = bvld[win] && (x >= qx1[win]) && (x < qx2[win]) &&
                   (y >= qy1[win]) && (y < qy2[win]);
        float mask = inw ? 1.0f : 0.0f;
        float ap = fminf(dl, dr) * fminf(dt, db)
                   / fmaxf(fmaxf(dl, dr), 1e-12f)
                   / fmaxf(fmaxf(dt, db), 1e-12f);
        float soft = inw ? ap * rawMeta[win] : 1.0f;
        const float s4 = 1.0f / (4.0f * stride);  // exact: 4*stride is a power of two
        res[tid][0] = dl * s4 * mask;
        res[tid][1] = dt * s4 * mask;
        res[tid][2] = dr * s4 * mask;
        res[tid][3] = db * s4 * mask;
        res[tid][4] = soft;
        res[tid][5] = mask;
        res[tid][6] = __int_as_float(blab[win]);
    }
    __syncthreads();

    // ---- Coalesced channel-last writes through the LDS transpose
    const size_t pixBase = (size_t)b * NPIX_IMG + (size_t)lvlOff + (size_t)block0;
    float* __restrict__ cls = cls_out + pixBase * CLS_CH;
    float* __restrict__ rg  = regr_out + pixBase * REG_CH;

    const int totC = P * CLS_CH;
    for (int i = tid; i < totC; i += 256) {
        int pix = i / CLS_CH;
        int ch  = i - pix * CLS_CH;
        float mask = res[pix][5];
        float v;
        if (ch < NUM_CLASSES)
            v = (ch == __float_as_int(res[pix][6])) ? mask : 0.0f;
        else if (ch == NUM_CLASSES)
            v = res[pix][4];
        else
            v = mask;
        cls[i] = v;
    }
    const int totR = P * REG_CH;
    for (int i = tid; i < totR; i += 256) {
        int pix = i / REG_CH;
        int ch  = i - pix * REG_CH;
        float v = (ch < 4) ? res[pix][ch] : ((ch == 4) ? res[pix][4] : res[pix][5]);
        rg[i] = v;
    }
}

extern "C" void kernel_launch(void* const* d_in, const int* in_sizes, int n_in,
                              void* d_out, int out_size, void* d_ws, size_t ws_size,
                              hipStream_t stream) {
    (void)in_sizes; (void)n_in; (void)out_size; (void)d_ws; (void)ws_size;
    // d_in[0] = fm_shapes (int32, static: 1024/{8,16,32,64,128}), unused
    const float* boxes = (const float*)d_in[1];  // (8,64,5)
    const float* meta  = (const float*)d_in[2];  // (8,64,5)
    float* out  = (float*)d_out;
    float* cls  = out;                                        // 8*21824*82 floats
    float* regr = out + (size_t)NIMG * NPIX_IMG * CLS_CH;     // 8*21824*6 floats

    struct Lv { int fh, fw; float stride; int off; };
    const Lv lv[5] = {
        {128, 128,   8.0f,     0},
        { 64,  64,  16.0f, 16384},
        { 32,  32,  32.0f, 20480},
        { 16,  16,  64.0f, 21504},
        {  8,   8, 128.0f, 21760},
    };
    for (int l = 0; l < 5; ++l) {
        const int npix = lv[l].fh * lv[l].fw;
        dim3 grid((npix + 255) / 256, NIMG);
        hipLaunchKernelGGL(sapd_level_kernel, grid, dim3(256), 0, stream,
                           boxes, meta, cls, regr,
                           l, lv[l].fh, lv[l].fw, lv[l].stride, lv[l].off);
    }
}